// KNN_18614388261211
// MI455X (gfx1250) — compile-verified
//
#include <hip/hip_runtime.h>

typedef __attribute__((ext_vector_type(2))) float v2f;
typedef __attribute__((ext_vector_type(8))) float v8f;

#define NPTS    16384      // 8 * 2048 points
#define KOUT    16
#define KK      17         // k+1 kept per row
#define SEGS    8          // waves per block = column segments
#define ROWSPB  32         // rows per block (2 WMMA row-tiles)
#define SEGCOLS (NPTS / SEGS)  // 2048 columns per wave

// D = A x B with A[i,:] = (-2xi,-2yi,-2zi, 1), B[:,j] = (xj,yj,zj,|xj|^2)
//  => D[i,j] = |xj|^2 - 2 xi.xj ;  dist^2 = D[i,j] + |xi|^2
__global__ __launch_bounds__(256) void knn_topk_wmma(const float* __restrict__ xg,
                                                     float* __restrict__ outD,
                                                     int*   __restrict__ outI)
{
    __shared__ float sTile[SEGS][ROWSPB][KK];  // per-wave 32x16 D tile (stride-17 pad)
    __shared__ float sMV[SEGS][ROWSPB][KK];    // per-segment top-17 values
    __shared__ int   sMI[SEGS][ROWSPB][KK];    // per-segment top-17 indices

    const int lane = threadIdx.x & 31;
    const int wave = threadIdx.x >> 5;
    const int hi   = lane >> 4;        // 0: lanes 0-15 (K=0,1), 1: lanes 16-31 (K=2,3)
    const int m    = lane & 15;
    const int rowBase = blockIdx.x * ROWSPB;

    // ---- Build A for the two 16-row tiles owned by this block (K=4 norm trick)
    const float* pa0 = xg + 3 * (rowBase + m);
    const float* pa1 = xg + 3 * (rowBase + 16 + m);
    float ax0 = pa0[0], ay0 = pa0[1], az0 = pa0[2];
    float ax1 = pa1[0], ay1 = pa1[1], az1 = pa1[2];
    v2f a0, a1;
    a0.x = hi ? (-2.0f * az0) : (-2.0f * ax0);
    a0.y = hi ? 1.0f          : (-2.0f * ay0);
    a1.x = hi ? (-2.0f * az1) : (-2.0f * ax1);
    a1.y = hi ? 1.0f          : (-2.0f * ay1);

    // ---- Each lane owns group-row == lane for the selection phase
    const float* pr = xg + 3 * (rowBase + lane);
    float rx = pr[0], ry = pr[1], rz = pr[2];
    const float sqr = rx * rx + ry * ry + rz * rz;

    float vals[KK]; int idxs[KK];
#pragma unroll
    for (int j = 0; j < KK; ++j) { vals[j] = -3.4e38f; idxs[j] = 0; }

    const int colBase = wave * SEGCOLS;

    for (int t = 0; t < SEGCOLS; t += 16) {
        const int col = colBase + t + m;
        const float* pb = xg + 3 * col;
        float bx = pb[0], by = pb[1], bz = pb[2];
        if (t + 16 < SEGCOLS) __builtin_prefetch(pb + 48, 0, 1);  // next tile
        float sqc = bx * bx + by * by + bz * bz;
        v2f b;
        b.x = hi ? bz  : bx;
        b.y = hi ? sqc : by;

        v8f c = {};
        v8f d0 = __builtin_amdgcn_wmma_f32_16x16x4_f32(false, a0, false, b,
                                                       (short)0, c, false, false);
        v8f d1 = __builtin_amdgcn_wmma_f32_16x16x4_f32(false, a1, false, b,
                                                       (short)0, c, false, false);

        // Scatter D tiles to LDS: group rows 0..31 x 16 cols, stride-17 padded
        const int rOff = hi ? 8 : 0;
#pragma unroll
        for (int j = 0; j < 8; ++j) {
            sTile[wave][rOff + j][m]      = d0[j];
            sTile[wave][16 + rOff + j][m] = d1[j];
        }
        // In-wave DS ordering: stores above land before the row loads below.

        // ---- Preload this lane's row into registers (contiguous -> merged
        //      ds_load_2addr pairs, ONE dscnt wait), then select on registers.
        float g[16];
#pragma unroll
        for (int n = 0; n < 16; ++n) g[n] = sTile[wave][lane][n];

        // Tile-level quick reject: only scan candidates if the best one
        // in this tile can enter the current top-17.
        float mx = g[0];
#pragma unroll
        for (int n = 1; n < 16; ++n) mx = fmaxf(mx, g[n]);

        if (mx + sqr > vals[KK - 1]) {
#pragma unroll
            for (int n = 0; n < 16; ++n) {
                float d2 = g[n] + sqr;                 // ||xi - xj||^2
                if (d2 > vals[KK - 1]) {
                    float _v = d2; int _i = colBase + t + n;
#pragma unroll
                    for (int j = 0; j < KK; ++j) {
                        bool sw = _v > vals[j];
                        float tv = vals[j]; int ti = idxs[j];
                        vals[j] = sw ? _v : tv;  idxs[j] = sw ? _i : ti;
                        _v = sw ? tv : _v;       _i = sw ? ti : _i;
                    }
                }
            }
        }
    }

    // ---- Publish per-segment lists, then merge in wave 0
#pragma unroll
    for (int j = 0; j < KK; ++j) { sMV[wave][lane][j] = vals[j]; sMI[wave][lane][j] = idxs[j]; }
    __syncthreads();

    if (wave == 0) {
        float fv[KK]; int fi[KK];
#pragma unroll
        for (int j = 0; j < KK; ++j) { fv[j] = -3.4e38f; fi[j] = 0; }
        for (int s = 0; s < SEGS; ++s) {
#pragma unroll
            for (int j = 0; j < KK; ++j) {
                float v = sMV[s][lane][j]; int id = sMI[s][lane][j];
                if (v > fv[KK - 1]) {
                    float _v = v; int _i = id;
#pragma unroll
                    for (int q = 0; q < KK; ++q) {
                        bool sw = _v > fv[q];
                        float tv = fv[q]; int ti = fi[q];
                        fv[q] = sw ? _v : tv; fi[q] = sw ? _i : ti;
                        _v = sw ? tv : _v;    _i = sw ? ti : _i;
                    }
                }
            }
        }
        const int row = rowBase + lane;
#pragma unroll
        for (int tq = 0; tq < KOUT; ++tq) {
            outD[row * KOUT + tq] = -fv[tq + 1];   // reference returns pair = -dist^2
            outI[row * KOUT + tq] = fi[tq + 1];    // index into flattened B*N
        }
    }
}

extern "C" void kernel_launch(void* const* d_in, const int* in_sizes, int n_in,
                              void* d_out, int out_size, void* d_ws, size_t ws_size,
                              hipStream_t stream) {
    (void)in_sizes; (void)n_in; (void)out_size; (void)d_ws; (void)ws_size;
    const float* x = (const float*)d_in[0];       // (8,2048,3) f32; d_in[1] = k (16, hardcoded)
    float* outD = (float*)d_out;                  // first 262144 floats: dists
    int*   outI = (int*)d_out + NPTS * KOUT;      // next 262144 ints: indices

    knn_topk_wmma<<<NPTS / ROWSPB, 256, 0, stream>>>(x, outD, outI);
}